// SaliencyLoss_14740327760077
// MI455X (gfx1250) — compile-verified
//
#include <hip/hip_runtime.h>

typedef float v2f __attribute__((ext_vector_type(2)));
typedef float v8f __attribute__((ext_vector_type(8)));

#define NB      16                 // batch rows
#define MROW    (512 * 512)        // elements per row
#define SPLITS  64                 // chunks per (loss,row) job in stats pass
#define CHUNK   (MROW / SPLITS)    // 4096 elements per block
#define NJOBS   (2 * NB)           // 2 losses x 16 rows
#define STW     4                  // stats stride per job: P, posSum, negSum, pad

// Sum all 16 distinct row-sums of a 16x16 f32 WMMA accumulator.
// D[m,n] is identical for every n (B was all-ones), so column n=0 suffices:
// VGPR r, lane 0 -> (M=r, N=0); lane 16 -> (M=r+8, N=0).
__device__ __forceinline__ float wave_wmma_total(v8f c) {
    float s = 0.f;
#pragma unroll
    for (int r = 0; r < 8; ++r) {
        s += __shfl(c[r], 0, 32);
        s += __shfl(c[r], 16, 32);
    }
    return s;
}

__global__ void __launch_bounds__(256) ohem_init(float* out, float* stats) {
    const int t = threadIdx.x;
    if (t < NJOBS * STW) stats[t] = 0.f;
    if (t == 0) out[0] = 0.f;
}

// ---------------------------------------------------------------------------
// Pass 1: streaming stats. For each (loss, row) job accumulate
//   P      = count(label >= 0.1)
//   posSum = sum(loss | pos),   negSum = sum(loss | !pos)
// The accumulation runs on the matrix pipe: D = A x ones(4x16) + C sums 64
// f32 values per v_wmma_f32_16x16x4_f32 with full f32 precision.
// ---------------------------------------------------------------------------
__global__ void __launch_bounds__(256) ohem_stats(
        const float* __restrict__ gh,  const float* __restrict__ gah,
        const float* __restrict__ pgh, const float* __restrict__ pgah,
        const float* __restrict__ mask, float* __restrict__ stats) {
    const int bid   = blockIdx.x;
    const int job   = bid >> 6;          // / SPLITS
    const int chunk = bid & (SPLITS - 1);
    const int li    = job >> 4;          // 0 = char(g), 1 = affinity(a)
    const int row   = job & (NB - 1);
    const float* __restrict__ lab = li ? gah  : gh;
    const float* __restrict__ prd = li ? pgah : pgh;
    const int base = row * MROW + chunk * CHUNK;
    const int tid  = threadIdx.x;

    v8f cPos = {}; v8f cNeg = {}; v8f cCnt = {};
    const v2f ones = {1.f, 1.f};

#pragma unroll
    for (int it = 0; it < 4; ++it) {
        const int e = base + it * 1024 + tid * 4;
        const float4 L  = *reinterpret_cast<const float4*>(lab  + e);
        const float4 Q  = *reinterpret_cast<const float4*>(prd  + e);
        const float4 Mk = *reinterpret_cast<const float4*>(mask + e);

        const float l0 = (Q.x - L.x) * (Q.x - L.x) * Mk.x;
        const float l1 = (Q.y - L.y) * (Q.y - L.y) * Mk.y;
        const float l2 = (Q.z - L.z) * (Q.z - L.z) * Mk.z;
        const float l3 = (Q.w - L.w) * (Q.w - L.w) * Mk.w;
        const bool p0 = (L.x >= 0.1f), p1 = (L.y >= 0.1f);
        const bool p2 = (L.z >= 0.1f), p3 = (L.w >= 0.1f);

        v2f aP0 = { p0 ? l0 : 0.f, p1 ? l1 : 0.f };
        v2f aN0 = { p0 ? 0.f : l0, p1 ? 0.f : l1 };
        v2f aC0 = { p0 ? 1.f : 0.f, p1 ? 1.f : 0.f };
        cPos = __builtin_amdgcn_wmma_f32_16x16x4_f32(false, aP0, false, ones, (short)0, cPos, false, false);
        cNeg = __builtin_amdgcn_wmma_f32_16x16x4_f32(false, aN0, false, ones, (short)0, cNeg, false, false);
        cCnt = __builtin_amdgcn_wmma_f32_16x16x4_f32(false, aC0, false, ones, (short)0, cCnt, false, false);

        v2f aP1 = { p2 ? l2 : 0.f, p3 ? l3 : 0.f };
        v2f aN1 = { p2 ? 0.f : l2, p3 ? 0.f : l3 };
        v2f aC1 = { p2 ? 1.f : 0.f, p3 ? 1.f : 0.f };
        cPos = __builtin_amdgcn_wmma_f32_16x16x4_f32(false, aP1, false, ones, (short)0, cPos, false, false);
        cNeg = __builtin_amdgcn_wmma_f32_16x16x4_f32(false, aN1, false, ones, (short)0, cNeg, false, false);
        cCnt = __builtin_amdgcn_wmma_f32_16x16x4_f32(false, aC1, false, ones, (short)0, cCnt, false, false);
    }

    const float sPos = wave_wmma_total(cPos);
    const float sNeg = wave_wmma_total(cNeg);
    const float sCnt = wave_wmma_total(cCnt);

    __shared__ float red[8][3];
    const int wave = tid >> 5, lane = tid & 31;
    if (lane == 0) { red[wave][0] = sPos; red[wave][1] = sNeg; red[wave][2] = sCnt; }
    __syncthreads();
    if (tid == 0) {
        float tP = 0.f, tN = 0.f, tC = 0.f;
#pragma unroll
        for (int w = 0; w < 8; ++w) { tP += red[w][0]; tN += red[w][1]; tC += red[w][2]; }
        float* st = stats + job * STW;
        atomicAdd(&st[0], tC);   // P (exact: < 2^24)
        atomicAdd(&st[1], tP);   // posSum
        atomicAdd(&st[2], tN);   // negSum
    }
}

// ---------------------------------------------------------------------------
// Pass 2: per-job finalize. Closed form when N <= 3P; otherwise exact top-k
// via 4x8-bit radix select on the loss bit patterns (loss >= 0, so uint order
// == float order) with LDS count/sum histograms. k = 3P (neg only) or 500
// (whole row, P == 0 fallback). Branch conditions are block-uniform.
// ---------------------------------------------------------------------------
__global__ void __launch_bounds__(256) ohem_finalize(
        const float* __restrict__ gh,  const float* __restrict__ gah,
        const float* __restrict__ pgh, const float* __restrict__ pgah,
        const float* __restrict__ mask, const float* __restrict__ stats,
        float* __restrict__ out) {
    __shared__ unsigned hCnt[256];
    __shared__ float    hSum[256];
    __shared__ float    fctl;     // running selected sum
    __shared__ unsigned uctl[3];  // remK, prefix, done

    const int job = blockIdx.x;
    const int li  = job >> 4;
    const int row = job & (NB - 1);
    const float* __restrict__ lab = li ? gah  : gh;
    const float* __restrict__ prd = li ? pgah : pgh;
    const int base = row * MROW;
    const int tid  = threadIdx.x;

    const float P      = stats[job * STW + 0];
    const float posSum = stats[job * STW + 1];
    const float negSum = stats[job * STW + 2];
    const float N      = (float)MROW - P;

    bool  needSelect = false, negOnly = false;
    int   k = 1;
    float per_row = 0.f;

    if (P > 0.5f) {
        const float posi = posSum / P;
        if (N < 0.5f) {                      // k=1, top sum of zeroed negs = 0
            per_row = posi;
        } else if (N <= 3.f * P) {           // k=N -> plain mean of negatives
            per_row = posi + negSum / N;
        } else {                             // k=3P -> true top-k needed
            needSelect = true; negOnly = true;
            k = 3 * (int)(P + 0.5f);
            per_row = posi;
        }
    } else {                                 // P==0 -> top-500 of whole row
        needSelect = true; negOnly = false; k = 500;
    }

    if (needSelect) {
        unsigned prefix = 0;
        int   bitsDone = 0, remK = k, done = 0;
        float sumSel = 0.f;
        for (int pass = 0; pass < 4 && !done; ++pass) {
            const int shift = 24 - 8 * pass;
            if (tid < 256) { hCnt[tid] = 0u; hSum[tid] = 0.f; }
            __syncthreads();
            for (int i = tid * 4; i < MROW; i += 256 * 4) {
                __builtin_prefetch(lab + base + i + 4096, 0, 1);
                const float4 L  = *reinterpret_cast<const float4*>(lab  + base + i);
                const float4 Q  = *reinterpret_cast<const float4*>(prd  + base + i);
                const float4 Mk = *reinterpret_cast<const float4*>(mask + base + i);
                const float lv[4] = {L.x, L.y, L.z, L.w};
                const float qv[4] = {Q.x, Q.y, Q.z, Q.w};
                const float mv[4] = {Mk.x, Mk.y, Mk.z, Mk.w};
#pragma unroll
                for (int j = 0; j < 4; ++j) {
                    const bool inS = negOnly ? (lv[j] < 0.1f) : true;
                    if (inS) {
                        const float d = qv[j] - lv[j];
                        const float v = d * d * mv[j];
                        const unsigned u = __float_as_uint(v);
                        const bool match = (bitsDone == 0) ||
                            ((u >> (32 - bitsDone)) == (prefix >> (32 - bitsDone)));
                        if (match) {
                            const int bin = (u >> shift) & 255;
                            atomicAdd(&hCnt[bin], 1u);
                            atomicAdd(&hSum[bin], v);
                        }
                    }
                }
            }
            __syncthreads();
            if (tid == 0) {
                int rk = remK; float ss = sumSel; unsigned pf = prefix; int dn = 0;
                for (int bin = 255; bin >= 0; --bin) {
                    const int c = (int)hCnt[bin];
                    if (c < rk)       { ss += hSum[bin]; rk -= c; }
                    else if (c == rk) { ss += hSum[bin]; rk = 0; dn = 1; break; }
                    else              { pf |= ((unsigned)bin) << shift; break; }
                }
                fctl = ss; uctl[0] = (unsigned)rk; uctl[1] = pf; uctl[2] = (unsigned)dn;
            }
            __syncthreads();
            sumSel = fctl; remK = (int)uctl[0]; prefix = uctl[1]; done = (int)uctl[2];
            bitsDone += 8;
        }
        if (!done && remK > 0)   // survivors all equal the fully-resolved value
            sumSel += (float)remK * __uint_as_float(prefix);

        if (negOnly) per_row += sumSel / (float)k;
        else         per_row  = sumSel / 500.f;
    }

    if (tid == 0) atomicAdd(out, per_row / (float)NB);
}

extern "C" void kernel_launch(void* const* d_in, const int* in_sizes, int n_in,
                              void* d_out, int out_size, void* d_ws, size_t ws_size,
                              hipStream_t stream) {
    const float* gh   = (const float*)d_in[0];
    const float* gah  = (const float*)d_in[1];
    const float* pgh  = (const float*)d_in[2];
    const float* pgah = (const float*)d_in[3];
    const float* mask = (const float*)d_in[4];
    float* out   = (float*)d_out;
    float* stats = (float*)d_ws;   // NJOBS * STW floats = 512 B

    ohem_init    <<<1,                 256, 0, stream>>>(out, stats);
    ohem_stats   <<<NJOBS * SPLITS,    256, 0, stream>>>(gh, gah, pgh, pgah, mask, stats);
    ohem_finalize<<<NJOBS,             256, 0, stream>>>(gh, gah, pgh, pgah, mask, stats, out);
}